// AttnDecoderRNN_84662395338799
// MI455X (gfx1250) — compile-verified
//
#include <hip/hip_runtime.h>
#include <math.h>

// ---- problem dims -----------------------------------------------------------
constexpr int BB = 64;        // batch
constexpr int LL = 2048;      // encoder length
constexpr int HH = 1024;      // hidden
constexpr int VV = 50257;     // vocab
constexpr int ABLK  = 4;              // attention blocks per batch (grid.x)
constexpr int PARTS = ABLK * 8;       // softmax partials per batch (8 waves/blk)
constexpr int LWAVE = LL / PARTS;     // l's handled per wave = 64

typedef __attribute__((ext_vector_type(2))) float v2f;
typedef __attribute__((ext_vector_type(8))) float v8f;

// ---- workspace layout (float offsets) --------------------------------------
constexpr long WS_GATES  = 0;                                   // B*4H
constexpr long WS_H1     = WS_GATES  + (long)BB * 4 * HH;       // B*H
constexpr long WS_CTX    = WS_H1     + (long)BB * HH;           // B*H
constexpr long WS_CONCAT = WS_CTX    + (long)BB * HH;           // B*H
constexpr long WS_ENERGY = WS_CONCAT + (long)BB * HH;           // B*L
constexpr long WS_PM     = WS_ENERGY + (long)BB * LL;           // B*PARTS
constexpr long WS_PS     = WS_PM     + (long)BB * PARTS;        // B*PARTS
constexpr long WS_PCTX   = WS_PS     + (long)BB * PARTS;        // B*PARTS*H
constexpr long WS_MS     = WS_PCTX   + (long)BB * PARTS * HH;   // 2*B

// fp32 WMMA helper: D = A(16x4) x B(4x16) + C, wave32
__device__ __forceinline__ v8f wmma_f32(v2f a, v2f b, v8f c) {
    return __builtin_amdgcn_wmma_f32_16x16x4_f32(false, a, false, b,
                                                 (short)0, c, false, false);
}

// ---------------------------------------------------------------------------
// Kernel 1: gates[64][4096] = emb[idx] @ W_ih^T + h0 @ W_hh^T (+ biases)
// One wave per 16-wide N tile; wave covers all 4 M-subtiles so weights are
// streamed exactly once.
// ---------------------------------------------------------------------------
__global__ __launch_bounds__(256) void k_gates(
    const int* __restrict__ idx, const float* __restrict__ emb,
    const float* __restrict__ h0, const float* __restrict__ W_ih,
    const float* __restrict__ W_hh, const float* __restrict__ b_ih,
    const float* __restrict__ b_hh, float* __restrict__ gates)
{
    const int wave = threadIdx.x >> 5;
    const int lane = threadIdx.x & 31;
    const int tile = blockIdx.x * 8 + wave;      // 256 tiles over N=4096
    const int n0   = tile * 16;
    const int nl   = lane & 15;
    const int half = lane >> 4;
    const int n    = n0 + nl;

    const float* xrow[4];
    const float* hrow[4];
#pragma unroll
    for (int t = 0; t < 4; ++t) {
        int m = t * 16 + nl;
        xrow[t] = emb + (long)idx[m] * HH;
        hrow[t] = h0  + (long)m * HH;
    }
    const float* wi = W_ih + (long)n * HH;
    const float* wh = W_hh + (long)n * HH;

    v8f c[4] = {};
    for (int k = 0; k < HH; k += 4) {
        const int ka = k + 2 * half;
        v2f bv = *reinterpret_cast<const v2f*>(wi + ka);
#pragma unroll
        for (int t = 0; t < 4; ++t) {
            v2f a = *reinterpret_cast<const v2f*>(xrow[t] + ka);
            c[t] = wmma_f32(a, bv, c[t]);
        }
    }
    for (int k = 0; k < HH; k += 4) {
        const int ka = k + 2 * half;
        v2f bv = *reinterpret_cast<const v2f*>(wh + ka);
#pragma unroll
        for (int t = 0; t < 4; ++t) {
            v2f a = *reinterpret_cast<const v2f*>(hrow[t] + ka);
            c[t] = wmma_f32(a, bv, c[t]);
        }
    }
    const float bias = b_ih[n] + b_hh[n];
#pragma unroll
    for (int t = 0; t < 4; ++t)
#pragma unroll
        for (int r = 0; r < 8; ++r) {
            int m = t * 16 + r + 8 * half;
            gates[(long)m * (4 * HH) + n] = c[t][r] + bias;
        }
}

// ---------------------------------------------------------------------------
// Kernel 2: LSTM pointwise (i,f,g,o) -> h1, c1
// ---------------------------------------------------------------------------
__global__ void k_lstm(const float* __restrict__ gates,
                       const float* __restrict__ c0,
                       float* __restrict__ h1_ws,
                       float* __restrict__ h1_out,
                       float* __restrict__ c1_out)
{
    int i = blockIdx.x * blockDim.x + threadIdx.x;   // B*H
    int b = i >> 10, h = i & (HH - 1);
    const float* g = gates + (long)b * 4 * HH;
    float gi = g[h], gf = g[HH + h], gg = g[2 * HH + h], go = g[3 * HH + h];
    float si = 1.f / (1.f + __expf(-gi));
    float sf = 1.f / (1.f + __expf(-gf));
    float so = 1.f / (1.f + __expf(-go));
    float tg = tanhf(gg);
    float c1 = sf * c0[i] + si * tg;
    float h1 = so * tanhf(c1);
    h1_ws[i]  = h1;
    h1_out[i] = h1;
    c1_out[i] = c1;
}

// ---------------------------------------------------------------------------
// Kernel 3: attention partials — flash-style online softmax.
// Each WAVE owns whole enc rows (lane covers 32 contiguous h via 8x b128
// loads), dot-product reduced with an intra-wave shfl_xor butterfly: no LDS,
// no barriers. Each wave is its own softmax partial (PARTS=32 per batch).
// encoder_outputs (536 MB) is read exactly once. grid = (ABLK, B).
// ---------------------------------------------------------------------------
__global__ __launch_bounds__(256) void k_attn_partial(
    const float* __restrict__ h1, const float* __restrict__ enc,
    float* __restrict__ energy, float* __restrict__ pm,
    float* __restrict__ ps, float* __restrict__ pctx)
{
    const int b    = blockIdx.y;
    const int blk  = blockIdx.x;          // 0..ABLK-1
    const int w    = threadIdx.x >> 5;    // wave in block
    const int lane = threadIdx.x & 31;

    // lane's h-slice: [lane*32, lane*32+32)
    const float* hb = h1 + (long)b * HH + lane * 32;
    float hv[32];
#pragma unroll
    for (int j = 0; j < 8; ++j) {
        float4 t = reinterpret_cast<const float4*>(hb)[j];
        hv[4*j+0] = t.x; hv[4*j+1] = t.y; hv[4*j+2] = t.z; hv[4*j+3] = t.w;
    }

    float ctx[32];
#pragma unroll
    for (int j = 0; j < 32; ++j) ctx[j] = 0.f;
    float m = -3.0e38f, s = 0.f;

    const int lbase = blk * (LL / ABLK) + w;     // this wave: lbase + 8*i
    for (int i = 0; i < LWAVE; ++i) {
        const int l = lbase + 8 * i;
        const float* e = enc + ((long)l * BB + b) * HH + lane * 32;
        float ev[32];
        float p = 0.f;
#pragma unroll
        for (int j = 0; j < 8; ++j) {
            float4 t = reinterpret_cast<const float4*>(e)[j];
            ev[4*j+0] = t.x; ev[4*j+1] = t.y; ev[4*j+2] = t.z; ev[4*j+3] = t.w;
            p += hv[4*j+0]*t.x + hv[4*j+1]*t.y + hv[4*j+2]*t.z + hv[4*j+3]*t.w;
        }
        // wave32 butterfly reduce: every lane ends with the full dot product
#pragma unroll
        for (int off = 16; off > 0; off >>= 1) p += __shfl_xor(p, off);

        if (lane == 0) energy[(long)b * LL + l] = p;

        float mn    = fmaxf(m, p);
        float scale = __expf(m - mn);
        float pw    = __expf(p - mn);
        s = s * scale + pw;
#pragma unroll
        for (int j = 0; j < 32; ++j) ctx[j] = ctx[j] * scale + pw * ev[j];
        m = mn;
    }

    const int pidx = b * PARTS + blk * 8 + w;
    if (lane == 0) { pm[pidx] = m; ps[pidx] = s; }
    float* pc = pctx + (long)pidx * HH + lane * 32;
#pragma unroll
    for (int j = 0; j < 8; ++j) {
        float4 t;
        t.x = ctx[4*j+0]; t.y = ctx[4*j+1]; t.z = ctx[4*j+2]; t.w = ctx[4*j+3];
        reinterpret_cast<float4*>(pc)[j] = t;
    }
}

// ---------------------------------------------------------------------------
// Kernel 4: combine PARTS partials per batch -> context[b][:], (m*, s*)
// ---------------------------------------------------------------------------
__global__ __launch_bounds__(256) void k_attn_combine(
    const float* __restrict__ pm, const float* __restrict__ ps,
    const float* __restrict__ pctx, float* __restrict__ ctx_out,
    float* __restrict__ msum)
{
    const int b = blockIdx.x, tid = threadIdx.x;
    float mstar = -3.0e38f;
#pragma unroll
    for (int i = 0; i < PARTS; ++i) mstar = fmaxf(mstar, pm[b * PARTS + i]);
    float w[PARTS];
    float sstar = 0.f;
#pragma unroll
    for (int i = 0; i < PARTS; ++i) {
        w[i] = __expf(pm[b * PARTS + i] - mstar);
        sstar += ps[b * PARTS + i] * w[i];
    }
    const float inv = 1.f / sstar;
#pragma unroll
    for (int j = 0; j < 4; ++j) {
        int h = tid + 256 * j;
        float acc = 0.f;
#pragma unroll
        for (int i = 0; i < PARTS; ++i)
            acc += pctx[(long)(b * PARTS + i) * HH + h] * w[i];
        ctx_out[b * HH + h] = acc * inv;
    }
    if (tid == 0) { msum[2 * b] = mstar; msum[2 * b + 1] = sstar; }
}

// ---------------------------------------------------------------------------
// Kernel 5: attn weights out = exp(e - m*)/s*
// ---------------------------------------------------------------------------
__global__ void k_attn_write(const float* __restrict__ energy,
                             const float* __restrict__ msum,
                             float* __restrict__ attn)
{
    int i = blockIdx.x * blockDim.x + threadIdx.x;   // B*L
    int b = i >> 11;
    attn[i] = __expf(energy[i] - msum[2 * b]) / msum[2 * b + 1];
}

// ---------------------------------------------------------------------------
// Kernel 6: concat_out = tanh([h1,ctx] @ W_concat^T + b_concat); WMMA fp32
// ---------------------------------------------------------------------------
__global__ __launch_bounds__(256) void k_concat(
    const float* __restrict__ h1, const float* __restrict__ ctx,
    const float* __restrict__ Wc, const float* __restrict__ bc,
    float* __restrict__ cout)
{
    const int wave = threadIdx.x >> 5;
    const int lane = threadIdx.x & 31;
    const int tile = blockIdx.x * 8 + wave;   // 64 tiles over N=1024
    const int n0   = tile * 16;
    const int nl   = lane & 15;
    const int half = lane >> 4;
    const int n    = n0 + nl;

    const float* a1[4];
    const float* a2[4];
#pragma unroll
    for (int t = 0; t < 4; ++t) {
        int m = t * 16 + nl;
        a1[t] = h1  + (long)m * HH;
        a2[t] = ctx + (long)m * HH;
    }
    const float* wr = Wc + (long)n * (2 * HH);

    v8f c[4] = {};
    for (int k = 0; k < HH; k += 4) {
        const int ka = k + 2 * half;
        v2f bv = *reinterpret_cast<const v2f*>(wr + ka);
#pragma unroll
        for (int t = 0; t < 4; ++t) {
            v2f a = *reinterpret_cast<const v2f*>(a1[t] + ka);
            c[t] = wmma_f32(a, bv, c[t]);
        }
    }
    for (int k = 0; k < HH; k += 4) {
        const int ka = k + 2 * half;
        v2f bv = *reinterpret_cast<const v2f*>(wr + HH + ka);
#pragma unroll
        for (int t = 0; t < 4; ++t) {
            v2f a = *reinterpret_cast<const v2f*>(a2[t] + ka);
            c[t] = wmma_f32(a, bv, c[t]);
        }
    }
    const float bias = bc[n];
#pragma unroll
    for (int t = 0; t < 4; ++t)
#pragma unroll
        for (int r = 0; r < 8; ++r) {
            int m = t * 16 + r + 8 * half;
            cout[(long)m * HH + n] = tanhf(c[t][r] + bias);
        }
}

// ---------------------------------------------------------------------------
// Kernel 7: output = concat_out @ W_out^T + b_out ; V=50257 (ragged last tile)
// W_out (206 MB) streamed exactly once: each wave covers all 4 M-subtiles.
// ---------------------------------------------------------------------------
__global__ __launch_bounds__(256) void k_out(
    const float* __restrict__ co, const float* __restrict__ Wo,
    const float* __restrict__ bo, float* __restrict__ out)
{
    const int NT   = (VV + 15) / 16;          // 3142
    const int wave = threadIdx.x >> 5;
    const int lane = threadIdx.x & 31;
    const int tile = blockIdx.x * 8 + wave;
    if (tile >= NT) return;                    // wave-uniform: EXEC stays full
    const int n0   = tile * 16;
    const int nl   = lane & 15;
    const int half = lane >> 4;
    const int n    = n0 + nl;
    const bool nok = n < VV;

    const float* wr = Wo + (long)(nok ? n : 0) * HH;
    const float* ar[4];
#pragma unroll
    for (int t = 0; t < 4; ++t) ar[t] = co + (long)(t * 16 + nl) * HH;

    v8f c[4] = {};
    for (int k = 0; k < HH; k += 4) {
        const int ka = k + 2 * half;
        v2f bv = *reinterpret_cast<const v2f*>(wr + ka);
#pragma unroll
        for (int t = 0; t < 4; ++t) {
            v2f a = *reinterpret_cast<const v2f*>(ar[t] + ka);
            c[t] = wmma_f32(a, bv, c[t]);
        }
    }
    if (nok) {
        const float bias = bo[n];
#pragma unroll
        for (int t = 0; t < 4; ++t)
#pragma unroll
            for (int r = 0; r < 8; ++r) {
                int m = t * 16 + r + 8 * half;
                out[(long)m * VV + n] = c[t][r] + bias;
            }
    }
}

// ---------------------------------------------------------------------------
extern "C" void kernel_launch(void* const* d_in, const int* in_sizes, int n_in,
                              void* d_out, int out_size, void* d_ws, size_t ws_size,
                              hipStream_t stream)
{
    const int*   input_seq = (const int*)  d_in[0];
    const float* enc       = (const float*)d_in[1];
    const float* h0        = (const float*)d_in[2];
    const float* c0        = (const float*)d_in[3];
    const float* emb       = (const float*)d_in[4];
    const float* W_ih      = (const float*)d_in[5];
    const float* W_hh      = (const float*)d_in[6];
    const float* b_ih      = (const float*)d_in[7];
    const float* b_hh      = (const float*)d_in[8];
    const float* W_concat  = (const float*)d_in[9];
    const float* b_concat  = (const float*)d_in[10];
    const float* W_out     = (const float*)d_in[11];
    const float* b_out     = (const float*)d_in[12];

    float* out      = (float*)d_out;
    float* out_h1   = out    + (long)BB * VV;
    float* out_c1   = out_h1 + (long)BB * HH;
    float* out_attn = out_c1 + (long)BB * HH;

    float* ws     = (float*)d_ws;
    float* gates  = ws + WS_GATES;
    float* h1     = ws + WS_H1;
    float* ctx    = ws + WS_CTX;
    float* conc   = ws + WS_CONCAT;
    float* energy = ws + WS_ENERGY;
    float* pm     = ws + WS_PM;
    float* ps     = ws + WS_PS;
    float* pctx   = ws + WS_PCTX;
    float* msum   = ws + WS_MS;

    k_gates<<<dim3((4 * HH / 16) / 8), dim3(256), 0, stream>>>(
        input_seq, emb, h0, W_ih, W_hh, b_ih, b_hh, gates);

    k_lstm<<<dim3(BB * HH / 256), dim3(256), 0, stream>>>(
        gates, c0, h1, out_h1, out_c1);

    k_attn_partial<<<dim3(ABLK, BB), dim3(256), 0, stream>>>(
        h1, enc, energy, pm, ps, pctx);

    k_attn_combine<<<dim3(BB), dim3(256), 0, stream>>>(
        pm, ps, pctx, ctx, msum);

    k_attn_write<<<dim3(BB * LL / 256), dim3(256), 0, stream>>>(
        energy, msum, out_attn);

    k_concat<<<dim3((HH / 16) / 8), dim3(256), 0, stream>>>(
        h1, ctx, W_concat, b_concat, conc);

    const int NT = (VV + 15) / 16;
    k_out<<<dim3((NT + 7) / 8), dim3(256), 0, stream>>>(
        conc, W_out, b_out, out);
}